// PolyActGCN_26551487824428
// MI455X (gfx1250) — compile-verified
//
#include <hip/hip_runtime.h>
#include <hip/hip_bf16.h>

#define D_FEAT 128
#define EPS_BN 1e-5f

typedef float v2f __attribute__((ext_vector_type(2)));
typedef float v8f __attribute__((ext_vector_type(8)));

// ---------------------------------------------------------------------------
// Zero a buffer (float4 granularity). n4 = number of float4 elements.
// ---------------------------------------------------------------------------
__global__ void pg_zero_f32(float* __restrict__ p, int n4) {
  int i = blockIdx.x * blockDim.x + threadIdx.x;
  if (i < n4) ((float4*)p)[i] = make_float4(0.f, 0.f, 0.f, 0.f);
}

// ---------------------------------------------------------------------------
// Initialize output to broadcast bias row: out[n*128 + d] = b[d].
// ---------------------------------------------------------------------------
__global__ void pg_init_bias(float* __restrict__ out, const float* __restrict__ b,
                             int n4) {
  int i = blockIdx.x * blockDim.x + threadIdx.x;
  if (i >= n4) return;
  int d = (i * 4) & (D_FEAT - 1);
  ((float4*)out)[i] = make_float4(b[d], b[d + 1], b[d + 2], b[d + 3]);
}

// ---------------------------------------------------------------------------
// H[N,128] = X[N,128] @ W[128,128] using V_WMMA_F32_16X16X4_F32.
// Block = 256 threads = 8 waves. Block handles a 16-row strip of X (staged in
// LDS, 8 KB); wave w computes the 16-column tile [w*16, w*16+16).
// A frag (16x4 f32): lanes 0-15 hold M=lane, K=kk+{0,1}; lanes 16-31 K=kk+{2,3}.
// B frag (4x16 f32): mirrored; VGPR0 = rows K=kk+0 / kk+2, VGPR1 = kk+1 / kk+3.
// C/D (16x16 f32): VGPR r -> row r (lanes 0-15) / row r+8 (lanes 16-31).
// ---------------------------------------------------------------------------
__global__ void pg_gemm_wmma(const float* __restrict__ X,
                             const float* __restrict__ W,
                             float* __restrict__ Hout, int N) {
  __shared__ float xs[16 * D_FEAT];  // 8 KB
  const int row0 = blockIdx.x * 16;

  // Cooperative load of the 16x128 X strip into LDS (float4 granularity).
  for (int i = threadIdx.x; i < 16 * D_FEAT / 4; i += blockDim.x) {
    int r = row0 + (i >> 5);  // 32 float4 per row
    float4 v = (r < N) ? ((const float4*)X)[(size_t)r * (D_FEAT / 4) + (i & 31)]
                       : make_float4(0.f, 0.f, 0.f, 0.f);
    ((float4*)xs)[i] = v;
  }
  __syncthreads();

  const int wave  = threadIdx.x >> 5;
  const int lane  = threadIdx.x & 31;
  const int col0  = wave * 16;
  const int mlane = lane & 15;
  const int khalf = (lane >> 4) * 2;  // 0 for lanes 0-15, 2 for lanes 16-31

  v8f acc = {};
  for (int kk = 0; kk < D_FEAT; kk += 4) {
    v2f a, b;
    const float* ap = xs + mlane * D_FEAT + kk + khalf;
    a.x = ap[0];
    a.y = ap[1];
    const float* bp = W + (size_t)(kk + khalf) * D_FEAT + col0 + mlane;
    b.x = bp[0];
    b.y = bp[D_FEAT];
    acc = __builtin_amdgcn_wmma_f32_16x16x4_f32(false, a, false, b, (short)0,
                                                acc, false, false);
  }

  const int mbase = (lane >> 4) * 8;
  float* out = Hout + (size_t)row0 * D_FEAT + col0 + mlane;
#pragma unroll
  for (int r = 0; r < 8; r++) {
    if (row0 + mbase + r < N) out[(size_t)(mbase + r) * D_FEAT] = acc[r];
  }
}

// ---------------------------------------------------------------------------
// Edge scatter: Agg[row[e]] += H[col[e]] * ew[e]. One wave32 per edge,
// each lane handles 4 features (float4 gather + 4x global_atomic_add_f32).
// ---------------------------------------------------------------------------
__global__ void pg_edge_scatter(const float* __restrict__ H,
                                const int* __restrict__ row,
                                const int* __restrict__ col,
                                const float* __restrict__ ew,
                                float* __restrict__ Agg, int E) {
  int gwave  = (blockIdx.x * blockDim.x + threadIdx.x) >> 5;
  int nwaves = (gridDim.x * blockDim.x) >> 5;
  int lane   = threadIdx.x & 31;
  for (int e = gwave; e < E; e += nwaves) {
    int r = row[e];
    int c = col[e];
    float w = ew[e];
    float4 v = ((const float4*)(H + (size_t)c * D_FEAT))[lane];
    float* dst = Agg + (size_t)r * D_FEAT + lane * 4;
    atomicAdd(dst + 0, v.x * w);
    atomicAdd(dst + 1, v.y * w);
    atomicAdd(dst + 2, v.z * w);
    atomicAdd(dst + 3, v.w * w);
  }
}

// ---------------------------------------------------------------------------
// Per-column sum & sum-of-squares over N rows. 128 threads/block (1 col each),
// each block covers ROWS rows, partials merged via f32 atomics into S[0..255].
// ---------------------------------------------------------------------------
#define STAT_ROWS 128
__global__ void pg_col_stats(const float* __restrict__ A, float* __restrict__ S,
                             int N) {
  int d  = threadIdx.x;  // 0..127
  int r0 = blockIdx.x * STAT_ROWS;
  int r1 = r0 + STAT_ROWS;
  if (r1 > N) r1 = N;
  float s = 0.f, sq = 0.f;
  for (int r = r0; r < r1; r++) {
    float v = A[(size_t)r * D_FEAT + d];
    s += v;
    sq = fmaf(v, v, sq);
  }
  atomicAdd(&S[d], s);
  atomicAdd(&S[D_FEAT + d], sq);
}

// ---------------------------------------------------------------------------
// Fused BatchNorm + quartic polynomial (Horner). Note: the GCN bias b cancels
// exactly inside BN (constant shift per column), so it is omitted upstream.
// ---------------------------------------------------------------------------
__global__ void pg_bn_poly(const float* __restrict__ A, const float* __restrict__ S,
                           const float* __restrict__ gamma,
                           const float* __restrict__ beta,
                           const float* __restrict__ c, float* __restrict__ Xo,
                           int n4, float invN) {
  int i = blockIdx.x * blockDim.x + threadIdx.x;
  if (i >= n4) return;
  float4 v = ((const float4*)A)[i];
  int dbase = (i * 4) & (D_FEAT - 1);
  float c0 = c[0], c1 = c[1], c2 = c[2], c3 = c[3], c4 = c[4];
  float in[4] = {v.x, v.y, v.z, v.w};
  float out[4];
#pragma unroll
  for (int j = 0; j < 4; j++) {
    int d = dbase + j;
    float mean = S[d] * invN;
    float var  = fmaf(-mean, mean, S[D_FEAT + d] * invN);
    float rs   = rsqrtf(var + EPS_BN);
    float t    = fmaf((in[j] - mean) * rs, gamma[d], beta[d]);
    float acc  = fmaf(c4, t, c3);
    acc = fmaf(acc, t, c2);
    acc = fmaf(acc, t, c1);
    acc = fmaf(acc, t, c0);
    out[j] = acc;
  }
  ((float4*)Xo)[i] = make_float4(out[0], out[1], out[2], out[3]);
}

// ---------------------------------------------------------------------------
// Host orchestration
// ---------------------------------------------------------------------------
extern "C" void kernel_launch(void* const* d_in, const int* in_sizes, int n_in,
                              void* d_out, int out_size, void* d_ws, size_t ws_size,
                              hipStream_t stream) {
  (void)n_in; (void)out_size; (void)ws_size;
  const float* nf   = (const float*)d_in[0];
  const int*   row  = (const int*)d_in[1];
  const int*   col  = (const int*)d_in[2];
  const float* ew   = (const float*)d_in[3];
  const float* W1   = (const float*)d_in[4];
  const float* W2   = (const float*)d_in[6];
  const float* W3   = (const float*)d_in[8];
  const float* b3   = (const float*)d_in[9];
  const float* g1   = (const float*)d_in[10];
  const float* be1  = (const float*)d_in[11];
  const float* g2   = (const float*)d_in[12];
  const float* be2  = (const float*)d_in[13];
  const float* coef = (const float*)d_in[14];  // [2,5] flat
  float* out = (float*)d_out;

  const int N = in_sizes[0] / D_FEAT;
  const int E = in_sizes[1];
  const size_t ND = (size_t)N * D_FEAT;
  const float invN = 1.0f / (float)N;

  float* H = (float*)d_ws;        // [N, D] gemm output / recycled
  float* A = H + ND;              // [N, D] aggregation / recycled
  float* S = A + ND;              // [2*D] column stats

  const int nd4 = (int)(ND / 4);
  const int gemm_grid = (N + 15) / 16;
  const int stat_grid = (N + STAT_ROWS - 1) / STAT_ROWS;
  const int zero_grid = (nd4 + 255) / 256;
  const int bn_grid   = (nd4 + 255) / 256;
  const int edge_grid = 4096;

  // ---------------- Layer 1 ----------------
  pg_gemm_wmma<<<gemm_grid, 256, 0, stream>>>(nf, W1, H, N);
  pg_zero_f32<<<zero_grid, 256, 0, stream>>>(A, nd4);
  pg_edge_scatter<<<edge_grid, 256, 0, stream>>>(H, row, col, ew, A, E);
  pg_zero_f32<<<1, 64, 0, stream>>>(S, 2 * D_FEAT / 4);
  pg_col_stats<<<stat_grid, D_FEAT, 0, stream>>>(A, S, N);
  pg_bn_poly<<<bn_grid, 256, 0, stream>>>(A, S, g1, be1, coef + 0, H, nd4, invN);

  // ---------------- Layer 2 ----------------
  pg_gemm_wmma<<<gemm_grid, 256, 0, stream>>>(H, W2, A, N);
  pg_zero_f32<<<zero_grid, 256, 0, stream>>>(H, nd4);
  pg_edge_scatter<<<edge_grid, 256, 0, stream>>>(A, row, col, ew, H, E);
  pg_zero_f32<<<1, 64, 0, stream>>>(S, 2 * D_FEAT / 4);
  pg_col_stats<<<stat_grid, D_FEAT, 0, stream>>>(H, S, N);
  pg_bn_poly<<<bn_grid, 256, 0, stream>>>(H, S, g2, be2, coef + 5, A, nd4, invN);

  // ---------------- Layer 3 ----------------
  pg_gemm_wmma<<<gemm_grid, 256, 0, stream>>>(A, W3, H, N);
  pg_init_bias<<<bn_grid, 256, 0, stream>>>(out, b3, nd4);
  pg_edge_scatter<<<edge_grid, 256, 0, stream>>>(H, row, col, ew, out, E);
}